// GraphCNN_multihead_bert_gate_cls_83751862272624
// MI455X (gfx1250) — compile-verified
//
#include <hip/hip_runtime.h>
#include <math.h>

#define NN    1024
#define Hdim  128
#define GHd   32
#define LNUM  4
#define NHEAD 4
#define DKd   32

typedef float v2f __attribute__((ext_vector_type(2)));
typedef float v8f __attribute__((ext_vector_type(8)));

// ---------------------------------------------------------------------------
// prep: c = Wh@w1 + Wt@w2 (for e_node), r = Wr@w3 (for e_edge)
// ---------------------------------------------------------------------------
__global__ void prep_kernel(const float* __restrict__ Wh, const float* __restrict__ Wt,
                            const float* __restrict__ Wr, const float* __restrict__ gat_w,
                            float* __restrict__ cvec, float* __restrict__ rvec)
{
    int k = threadIdx.x;
    const float* w1 = gat_w;
    const float* w2 = gat_w + Hdim;
    const float* w3 = gat_w + 2 * Hdim;
    float sc = 0.f, sr = 0.f;
    for (int j = 0; j < Hdim; ++j) {
        sc += Wh[k * Hdim + j] * w1[j] + Wt[k * Hdim + j] * w2[j];
        sr += Wr[k * Hdim + j] * w3[j];
    }
    cvec[k] = sc;
    rvec[k] = sr;
}

// ---------------------------------------------------------------------------
// e_node[i] = node_feat[i,:] . cvec   (one wave per row, wave32 shuffle reduce)
// ---------------------------------------------------------------------------
__global__ void enode_kernel(const float* __restrict__ x, const float* __restrict__ cvec,
                             float* __restrict__ e_node)
{
    int wave = threadIdx.x >> 5, lane = threadIdx.x & 31;
    int i = blockIdx.x * 8 + wave;
    float4 xv = ((const float4*)(x + (size_t)i * Hdim))[lane];
    float4 cv = ((const float4*)cvec)[lane];
    float d = xv.x * cv.x + xv.y * cv.y + xv.z * cv.z + xv.w * cv.w;
    for (int off = 16; off; off >>= 1) d += __shfl_xor(d, off, 32);
    if (lane == 0) e_node[i] = d;
}

// ---------------------------------------------------------------------------
// Fused single pass over 512MB edge_feat (the roofline term: ~22us @ 23.3TB/s):
//   e_edge[i,j]    = edge_feat[i,j,:] . r
//   edge_mean[i,k] = mean_j edge_feat[i,j,k]
// One block (8 waves) per row i; each wave strides over j; b128 loads per lane.
// ---------------------------------------------------------------------------
__global__ void edge_pass_kernel(const float* __restrict__ edge_feat,
                                 const float* __restrict__ rvec,
                                 float* __restrict__ e_edge,
                                 float* __restrict__ edge_mean)
{
    int wave = threadIdx.x >> 5, lane = threadIdx.x & 31;
    size_t i = blockIdx.x;
    const float4* rowbase = (const float4*)(edge_feat + i * (size_t)NN * Hdim);
    float4 r4 = ((const float4*)rvec)[lane];
    float4 acc = make_float4(0.f, 0.f, 0.f, 0.f);
    for (int j = wave; j < NN; j += 8) {
        float4 v = rowbase[(size_t)j * 32 + lane];
        if (j + 8 < NN) __builtin_prefetch(&rowbase[(size_t)(j + 8) * 32 + lane], 0, 0);
        acc.x += v.x; acc.y += v.y; acc.z += v.z; acc.w += v.w;
        float d = v.x * r4.x + v.y * r4.y + v.z * r4.z + v.w * r4.w;
        for (int off = 16; off; off >>= 1) d += __shfl_xor(d, off, 32);
        if (lane == 0) e_edge[i * NN + j] = d;
    }
    __shared__ float smem[8 * 128];
    smem[wave * 128 + lane * 4 + 0] = acc.x;
    smem[wave * 128 + lane * 4 + 1] = acc.y;
    smem[wave * 128 + lane * 4 + 2] = acc.z;
    smem[wave * 128 + lane * 4 + 3] = acc.w;
    __syncthreads();
    int t = threadIdx.x;
    if (t < 128) {
        float s = 0.f;
#pragma unroll
        for (int w = 0; w < 8; ++w) s += smem[w * 128 + t];
        edge_mean[i * Hdim + t] = s * (1.0f / NN);
    }
}

// ---------------------------------------------------------------------------
// Row softmax in place: row[j] = softmax_j( (row[j] + addcol[j]) * scale )
// ---------------------------------------------------------------------------
__global__ void softmax_rows(float* __restrict__ mat, const float* __restrict__ addcol,
                             float scale, int ncols)
{
    __shared__ float red[256];
    size_t i = blockIdx.x;
    float* row = mat + i * ncols;
    int t = threadIdx.x;
    float mx = -3.4e38f;
    for (int j = t; j < ncols; j += 256) {
        float v = row[j] + (addcol ? addcol[j] : 0.f);
        mx = fmaxf(mx, v * scale);
    }
    red[t] = mx; __syncthreads();
    for (int s = 128; s; s >>= 1) { if (t < s) red[t] = fmaxf(red[t], red[t + s]); __syncthreads(); }
    mx = red[0]; __syncthreads();
    float sum = 0.f;
    for (int j = t; j < ncols; j += 256) {
        float v = row[j] + (addcol ? addcol[j] : 0.f);
        v = __expf(v * scale - mx);
        row[j] = v;
        sum += v;
    }
    red[t] = sum; __syncthreads();
    for (int s = 128; s; s >>= 1) { if (t < s) red[t] += red[t + s]; __syncthreads(); }
    float inv = 1.f / red[0];
    for (int j = t; j < ncols; j += 256) row[j] *= inv;
}

// ---------------------------------------------------------------------------
// General fp32 WMMA GEMM: C = A(MxK) @ B(KxNc) [+bias per col][+addmat][relu]
// One wave per 16x16 tile, 4 waves/block along M. K % 16 == 0, M,Nc % 16 == 0.
// TRANSB is a template parameter -> straight-line inner loop, pointer
// induction, 4x unrolled V_WMMA_F32_16X16X4_F32 per trip.
// ---------------------------------------------------------------------------
template <int TRANSB>
__global__ void gemm_wmma_f32(const float* __restrict__ A, int lda,
                              const float* __restrict__ B, int ldb,
                              float* __restrict__ C, int ldc,
                              int M, int K, int Nc,
                              const float* __restrict__ bias,
                              const float* __restrict__ addmat, int ldadd,
                              int relu)
{
    int wave = threadIdx.x >> 5;
    int lane = threadIdx.x & 31;
    int tn = blockIdx.x;
    int tm = blockIdx.y * 4 + wave;
    if (tm * 16 >= M) return;            // wave-uniform; EXEC stays full for WMMA

    int m    = tm * 16 + (lane & 15);
    int n    = tn * 16 + (lane & 15);
    int ksel = (lane >> 4) * 2;          // lanes 0-15: K=0,1 ; lanes 16-31: K=2,3

    const float* Ap = A + (size_t)m * lda + ksel;
    const float* Bp = TRANSB ? (B + (size_t)n * ldb + ksel)
                             : (B + (size_t)ksel * ldb + n);
    const size_t bstep = TRANSB ? 4 : (size_t)4 * ldb;

    v8f acc = {};
    for (int k0 = 0; k0 < K; k0 += 16) {
#pragma unroll
        for (int u = 0; u < 4; ++u) {
            v2f a = *(const v2f*)Ap;
            v2f b;
            if (TRANSB) {
                b = *(const v2f*)Bp;
            } else {
                b.x = Bp[0];
                b.y = Bp[ldb];
            }
            acc = __builtin_amdgcn_wmma_f32_16x16x4_f32(
                false, a, false, b, (short)0, acc, false, false);
            Ap += 4;
            Bp += bstep;
        }
    }

    // C/D layout: VGPR g -> (M = g + (lane>=16 ? 8 : 0), N = lane&15) within tile
    int rbase = tm * 16 + ((lane >> 4) * 8);
#pragma unroll
    for (int g = 0; g < 8; ++g) {
        int row = rbase + g;
        float v = acc[g];
        if (bias)   v += bias[n];
        if (addmat) v += addmat[(size_t)row * ldadd + n];
        if (relu)   v = fmaxf(v, 0.f);
        C[(size_t)row * ldc + n] = v;
    }
}

// ---------------------------------------------------------------------------
// O[i, :W] = A[i, :W] (+ Bm[i, :W])   with independent leading dims
// ---------------------------------------------------------------------------
__global__ void addcopy_kernel(const float* __restrict__ A, int lda,
                               const float* __restrict__ Bm, int ldb,
                               float* __restrict__ O, int ldo, int Wd)
{
    int idx = blockIdx.x * blockDim.x + threadIdx.x;
    if (idx >= NN * Wd) return;
    int i = idx / Wd, c = idx % Wd;
    float v = A[(size_t)i * lda + c];
    if (Bm) v += Bm[(size_t)i * ldb + c];
    O[(size_t)i * ldo + c] = v;
}

// ---------------------------------------------------------------------------
static inline void launch_gemm(hipStream_t s, const float* A, int lda,
                               const float* B, int ldb, int tB,
                               float* C, int ldc, int M, int K, int Nc,
                               const float* bias, const float* addmat, int ldadd, int relu)
{
    dim3 grid(Nc / 16, (M / 16 + 3) / 4);
    if (tB)
        gemm_wmma_f32<1><<<grid, 128, 0, s>>>(A, lda, B, ldb, C, ldc, M, K, Nc,
                                              bias, addmat, ldadd, relu);
    else
        gemm_wmma_f32<0><<<grid, 128, 0, s>>>(A, lda, B, ldb, C, ldc, M, K, Nc,
                                              bias, addmat, ldadd, relu);
}

extern "C" void kernel_launch(void* const* d_in, const int* in_sizes, int n_in,
                              void* d_out, int out_size, void* d_ws, size_t ws_size,
                              hipStream_t stream)
{
    const float* node_feat  = (const float*)d_in[0];
    const float* edge_feat  = (const float*)d_in[1];
    const float* gat_Wh     = (const float*)d_in[2];
    const float* gat_Wt     = (const float*)d_in[4];
    const float* gat_Wr     = (const float*)d_in[6];
    const float* gat_w      = (const float*)d_in[8];
    const float* gc1_We     = (const float*)d_in[10];
    const float* gc1_Wn[4]  = {(const float*)d_in[11], (const float*)d_in[12],
                               (const float*)d_in[13], (const float*)d_in[14]};
    const float* gc1_lin_W  = (const float*)d_in[15];
    const float* gc1_lin_b  = (const float*)d_in[16];
    const float* mha_Wq     = (const float*)d_in[17];
    const float* mha_bq     = (const float*)d_in[18];
    const float* mgc_We     = (const float*)d_in[19];
    const float* mgc_Wn[4]  = {(const float*)d_in[20], (const float*)d_in[21],
                               (const float*)d_in[22], (const float*)d_in[23]};
    const float* mgc_lin_W  = (const float*)d_in[24];
    const float* mgc_lin_b  = (const float*)d_in[25];

    // workspace layout (floats)
    float* W         = (float*)d_ws;
    float* cvec      = W;                                   // 128
    float* rvec      = cvec + 128;                          // 128
    float* e_node    = rvec + 128;                          // 1024
    float* edge_mean = e_node + NN;                         // N*H
    float* adj       = edge_mean + (size_t)NN * Hdim;       // N*N
    float* att       = adj + (size_t)NN * NN;               // 4*N*N
    float* qbuf      = att + (size_t)NHEAD * NN * NN;       // 4*N*DK
    float* Xcat      = qbuf + (size_t)NHEAD * NN * DKd;     // N*256
    float* EW        = Xcat + (size_t)NN * 256;             // N*GH
    float* x1pre     = EW + (size_t)NN * GHd;               // N*H
    float* x1        = x1pre + (size_t)NN * Hdim;           // N*H
    float* Hout      = x1 + (size_t)NN * Hdim;              // N*512
    float* Tbuf      = Hout + (size_t)NN * (NHEAD * Hdim);  // N*GH

    // 1) folded GAT vectors (bias/scalar terms are softmax-invariant and dropped)
    prep_kernel<<<1, 128, 0, stream>>>(gat_Wh, gat_Wt, gat_Wr, gat_w, cvec, rvec);
    enode_kernel<<<NN / 8, 256, 0, stream>>>(node_feat, cvec, e_node);

    // 2) single fused pass over the 512MB edge tensor (bandwidth-bound)
    edge_pass_kernel<<<NN, 256, 0, stream>>>(edge_feat, rvec, adj, edge_mean);

    // 3) adj0 = softmax(e_edge + e_node[col]) ; rows sum to 1 => deg division skipped
    softmax_rows<<<NN, 256, 0, stream>>>(adj, e_node, 1.0f, NN);

    // 4) gc_block #1: Xcat = [node_feat | g1 | g2 | g3 | g4]
    addcopy_kernel<<<(NN * Hdim + 255) / 256, 256, 0, stream>>>(
        node_feat, Hdim, nullptr, 0, Xcat, 256, Hdim);
    for (int l = 0; l < LNUM; ++l) {
        int Kx = Hdim + GHd * l;
        launch_gemm(stream, edge_mean, Hdim, gc1_We + (size_t)l * Hdim * GHd, GHd, 0,
                    EW, GHd, NN, Hdim, GHd, nullptr, nullptr, 0, 0);
        launch_gemm(stream, Xcat, 256, gc1_Wn[l], GHd, 0,
                    Tbuf, GHd, NN, Kx, GHd, nullptr, nullptr, 0, 0);
        launch_gemm(stream, adj, NN, Tbuf, GHd, 0,
                    Xcat + Hdim + GHd * l, 256, NN, NN, GHd, nullptr, EW, GHd, 1);
    }
    // x1 = (concat(g) + x0) @ lin_W + lin_b
    addcopy_kernel<<<(NN * Hdim + 255) / 256, 256, 0, stream>>>(
        Xcat + Hdim, 256, node_feat, Hdim, x1pre, Hdim, Hdim);
    launch_gemm(stream, x1pre, Hdim, gc1_lin_W, Hdim, 0,
                x1, Hdim, NN, Hdim, Hdim, gc1_lin_b, nullptr, 0, 0);

    // 5) MHA attention maps: q_h = x1 @ Wq_h + bq_h ; att_h = softmax(q q^T / sqrt(DK))
    for (int h = 0; h < NHEAD; ++h) {
        launch_gemm(stream, x1, Hdim, mha_Wq + (size_t)h * Hdim * DKd, DKd, 0,
                    qbuf + (size_t)h * NN * DKd, DKd, NN, Hdim, DKd,
                    mha_bq + h * DKd, nullptr, 0, 0);
    }
    for (int h = 0; h < NHEAD; ++h) {
        float* att_h = att + (size_t)h * NN * NN;
        const float* q_h = qbuf + (size_t)h * NN * DKd;
        launch_gemm(stream, q_h, DKd, q_h, DKd, 1,
                    att_h, NN, NN, DKd, NN, nullptr, nullptr, 0, 0);
        softmax_rows<<<NN, 256, 0, stream>>>(att_h, nullptr, 0.17677669529663689f, NN);
    }

    // 6) per-head gc_blocks on x1, outputs concatenated into Hout
    for (int h = 0; h < NHEAD; ++h) {
        float* att_h = att + (size_t)h * NN * NN;
        addcopy_kernel<<<(NN * Hdim + 255) / 256, 256, 0, stream>>>(
            x1, Hdim, nullptr, 0, Xcat, 256, Hdim);
        for (int l = 0; l < LNUM; ++l) {
            int Kx = Hdim + GHd * l;
            const float* We_hl = mgc_We + (size_t)(h * LNUM + l) * Hdim * GHd;
            const float* Wn_hl = mgc_Wn[l] + (size_t)h * Kx * GHd;
            launch_gemm(stream, edge_mean, Hdim, We_hl, GHd, 0,
                        EW, GHd, NN, Hdim, GHd, nullptr, nullptr, 0, 0);
            launch_gemm(stream, Xcat, 256, Wn_hl, GHd, 0,
                        Tbuf, GHd, NN, Kx, GHd, nullptr, nullptr, 0, 0);
            launch_gemm(stream, att_h, NN, Tbuf, GHd, 0,
                        Xcat + Hdim + GHd * l, 256, NN, NN, GHd, nullptr, EW, GHd, 1);
        }
        addcopy_kernel<<<(NN * Hdim + 255) / 256, 256, 0, stream>>>(
            Xcat + Hdim, 256, x1, Hdim, Hout + (size_t)h * Hdim, NHEAD * Hdim, Hdim);
    }

    // 7) out = Hout @ mgc_lin_W + mgc_lin_b
    launch_gemm(stream, Hout, NHEAD * Hdim, mgc_lin_W, Hdim, 0,
                (float*)d_out, Hdim, NN, NHEAD * Hdim, Hdim,
                mgc_lin_b, nullptr, 0, 0);
}